// STFLayer_16183436771480
// MI455X (gfx1250) — compile-verified
//
#include <hip/hip_runtime.h>

// ---------------------------------------------------------------------------
// Multi-scale STFT layer on gfx1250 (MI455X), fp32 WMMA (V_WMMA_F32_16X16X4_F32)
// Scales n = {64,128,256}; per scale:
//   P[f,c]   = DFT of frame   (WMMA GEMM, A = trig table shared across frames)
//   O[f,o]   = P . K_f        (per-frequency resized filter, VALU via LDS)
//   act      = leaky_relu on Re/Im
//   y[t,o]   = iDFT of O      (WMMA GEMM) + bias, scattered into out[B,T,96]
// WMMA loops are software-pipelined: operands for step k+1 are fetched before
// the matrix ops of step k so s_wait_loadcnt need not drain to zero.
// ---------------------------------------------------------------------------

typedef float v2f __attribute__((ext_vector_type(2)));
typedef float v8f __attribute__((ext_vector_type(8)));

#define LRELU_ALPHA 0.2f
#define TWO_PI 6.28318530717958647692f

__device__ __forceinline__ v8f wmma4(v2f a, v2f b, v8f c) {
  // D = A(16x4 f32) * B(4x16 f32) + C(16x16 f32)
  return __builtin_amdgcn_wmma_f32_16x16x4_f32(false, a, false, b,
                                               (short)0, c, false, false);
}

// ---------------------------------------------------------------------------
// Init: trig tables + bilinear-resized complex filters (per scale).
//   Wr/Wi : [fep][n]   forward DFT rows (cos, -sin), zero for f >= fe
//   Cr/Ci : [n][fep]   inverse rows  a_f*cos/n, -a_f*sin/n, zero pad cols
//   Kr/Ki : [fe][2048] resize(kernel, 65 -> fe) along f (half-pixel bilinear)
// ---------------------------------------------------------------------------
__global__ void init_tables(const float* __restrict__ kr_in,
                            const float* __restrict__ ki_in,
                            float* __restrict__ Wr, float* __restrict__ Wi,
                            float* __restrict__ Cr, float* __restrict__ Ci,
                            float* __restrict__ Kr, float* __restrict__ Ki,
                            int n, int fe, int fep) {
  const int total_w = fep * n;
  const int total_c = n * fep;
  const int total_k = fe * 2048;
  const int total = total_w + total_c + total_k;
  const float inv_n = 1.0f / (float)n;
  for (int idx = blockIdx.x * blockDim.x + threadIdx.x; idx < total;
       idx += gridDim.x * blockDim.x) {
    if (idx < total_w) {
      int f = idx / n, t = idx - f * n;
      float c = 0.0f, s = 0.0f;
      if (f < fe) {
        float ang = TWO_PI * (float)((f * t) % n) * inv_n;
        sincosf(ang, &s, &c);
      }
      Wr[idx] = (f < fe) ? c : 0.0f;
      Wi[idx] = (f < fe) ? -s : 0.0f;
    } else if (idx < total_w + total_c) {
      int r = idx - total_w;
      int t = r / fep, f = r - t * fep;
      if (f < fe) {
        float a = (f == 0 || f == (n >> 1)) ? 1.0f : 2.0f;
        float ang = TWO_PI * (float)((f * t) % n) * inv_n;
        float s, c;
        sincosf(ang, &s, &c);
        Cr[r] = a * c * inv_n;
        Ci[r] = -a * s * inv_n;
      } else {
        Cr[r] = 0.0f;
        Ci[r] = 0.0f;
      }
    } else {
      int r = idx - total_w - total_c;
      int f = r >> 11, e = r & 2047;
      float scale = 65.0f / (float)fe;
      float pos = ((float)f + 0.5f) * scale - 0.5f;
      int lo = (int)floorf(pos);
      if (lo < 0) lo = 0;
      if (lo > 64) lo = 64;
      int hi = lo + 1;
      if (hi > 64) hi = 64;
      float w = pos - (float)lo;
      w = fminf(fmaxf(w, 0.0f), 1.0f);
      Kr[f * 2048 + e] = kr_in[lo * 2048 + e] * (1.0f - w) + kr_in[hi * 2048 + e] * w;
      Ki[f * 2048 + e] = ki_in[lo * 2048 + e] * (1.0f - w) + ki_in[hi * 2048 + e] * w;
    }
  }
}

// ---------------------------------------------------------------------------
// Stage A+B: per (frame, 16-frequency tile).
//   4 waves: wave w owns channel tile c = [16w,16w+16). WMMA over K = n (t).
//   P tile (16f x 64c, complex) -> LDS -> per-(f,o) complex channel mix with
//   the per-frequency resized filter, leaky-relu, write O[frame][fep][32].
// ---------------------------------------------------------------------------
__global__ void __launch_bounds__(128)
stage_ab(const float* __restrict__ x,
         const float* __restrict__ Wr, const float* __restrict__ Wi,
         const float* __restrict__ Kr, const float* __restrict__ Ki,
         float* __restrict__ Ore, float* __restrict__ Oim,
         int n, int fe, int fep) {
  __shared__ float LPr[16][65];
  __shared__ float LPi[16][65];

  const int frame = blockIdx.y;
  const int ftile = blockIdx.x;
  const int tid = threadIdx.x;
  const int lane = tid & 31;
  const int wave = tid >> 5;
  const int col = lane & 15;  // A: M index / B,C: N index
  const int kh = lane >> 4;   // which K half of the 16x4 / 4x16 operand

  const float* xb = x + (size_t)frame * n * 64;
  const float* wr_row = Wr + (size_t)(ftile * 16 + col) * n;
  const float* wi_row = Wi + (size_t)(ftile * 16 + col) * n;
  const int c_col = wave * 16 + col;

  auto ld_op = [&](int t0, v2f& ar, v2f& ai, v2f& bx) {
    ar.x = wr_row[t0 + 2 * kh];
    ar.y = wr_row[t0 + 2 * kh + 1];
    ai.x = wi_row[t0 + 2 * kh];
    ai.y = wi_row[t0 + 2 * kh + 1];
    bx.x = xb[(t0 + 2 * kh) * 64 + c_col];
    bx.y = xb[(t0 + 2 * kh + 1) * 64 + c_col];
  };

  v8f accR = {};
  v8f accI = {};
  v2f ar, ai, bx;
  ld_op(0, ar, ai, bx);
  for (int t0 = 4; t0 < n; t0 += 4) {
    v2f par = ar, pai = ai, pbx = bx;
    ld_op(t0, ar, ai, bx);            // prefetch next step before matrix ops
    accR = wmma4(par, pbx, accR);     // Pr += cos . x
    accI = wmma4(pai, pbx, accI);     // Pi += -sin . x
  }
  accR = wmma4(ar, bx, accR);
  accI = wmma4(ai, bx, accI);

#pragma unroll
  for (int r = 0; r < 8; ++r) {
    LPr[r + 8 * kh][wave * 16 + col] = accR[r];
    LPi[r + 8 * kh][wave * 16 + col] = accI[r];
  }
  __syncthreads();

  // per-frequency complex channel mix (64 -> 32) + leaky relu
#pragma unroll
  for (int q = 0; q < 4; ++q) {
    int idx = q * 128 + tid;  // 512 = 16 f x 32 o
    int fl = idx >> 5;
    int o = idx & 31;
    int f = ftile * 16 + fl;
    size_t obase = ((size_t)frame * fep + f) * 32 + o;
    if (f < fe) {
      const float* krow = Kr + (size_t)f * 2048 + o;
      const float* kirow = Ki + (size_t)f * 2048 + o;
      float orr = 0.0f, oii = 0.0f;
      for (int c = 0; c < 64; ++c) {
        float pr = LPr[fl][c];
        float pi = LPi[fl][c];
        float kre = krow[c * 32];
        float kim = kirow[c * 32];
        orr += pr * kre - pi * kim;
        oii += pr * kim + pi * kre;
      }
      orr = (orr >= 0.0f) ? orr : LRELU_ALPHA * orr;
      oii = (oii >= 0.0f) ? oii : LRELU_ALPHA * oii;
      Ore[obase] = orr;
      Oim[obase] = oii;
    } else {
      Ore[obase] = 0.0f;  // zero pad rows so stage D can run over fep blindly
      Oim[obase] = 0.0f;
    }
  }
}

// ---------------------------------------------------------------------------
// Stage D: per frame, y[n x 32] = Cr[n x fep].Or + Ci[n x fep].Oi  (WMMA),
// + bias, scattered into out[b, s*n + t, 32*j + o].
// 8 waves; tiles = (n/16) x 2 distributed round-robin (uniform trip count).
// ---------------------------------------------------------------------------
__global__ void __launch_bounds__(256)
stage_d(const float* __restrict__ Cr, const float* __restrict__ Ci,
        const float* __restrict__ Ore, const float* __restrict__ Oim,
        const float* __restrict__ bias, float* __restrict__ out,
        int n, int fep, int jch) {
  const int frame = blockIdx.x;
  const int lane = threadIdx.x & 31;
  const int wave = threadIdx.x >> 5;
  const int col = lane & 15;
  const int kh = lane >> 4;
  const int ntiles = (n >> 4) * 2;  // 8, 16 or 32 -> multiple of 8

  const float* ob = Ore + (size_t)frame * fep * 32;
  const float* oib = Oim + (size_t)frame * fep * 32;

  for (int tile = wave; tile < ntiles; tile += 8) {
    const int mt = tile >> 1;  // time tile
    const int nt = tile & 1;   // out-channel tile
    const float* cr_row = Cr + (size_t)(mt * 16 + col) * fep;
    const float* ci_row = Ci + (size_t)(mt * 16 + col) * fep;
    const int ocol = nt * 16 + col;

    auto ld_op = [&](int k0, v2f& acr, v2f& aci, v2f& bor, v2f& boi) {
      acr.x = cr_row[k0 + 2 * kh];
      acr.y = cr_row[k0 + 2 * kh + 1];
      aci.x = ci_row[k0 + 2 * kh];
      aci.y = ci_row[k0 + 2 * kh + 1];
      bor.x = ob[(k0 + 2 * kh) * 32 + ocol];
      bor.y = ob[(k0 + 2 * kh + 1) * 32 + ocol];
      boi.x = oib[(k0 + 2 * kh) * 32 + ocol];
      boi.y = oib[(k0 + 2 * kh + 1) * 32 + ocol];
    };

    v8f acc = {};
    v2f acr, aci, bor, boi;
    ld_op(0, acr, aci, bor, boi);
    for (int k0 = 4; k0 < fep; k0 += 4) {
      v2f pcr = acr, pci = aci, por = bor, poi = boi;
      ld_op(k0, acr, aci, bor, boi);  // prefetch next step before matrix ops
      acc = wmma4(pcr, por, acc);
      acc = wmma4(pci, poi, acc);
    }
    acc = wmma4(acr, bor, acc);
    acc = wmma4(aci, boi, acc);

    const int ch = jch + nt * 16 + col;
    const float bv = bias[ch];
#pragma unroll
    for (int r = 0; r < 8; ++r) {
      int t = mt * 16 + r + 8 * kh;
      out[((size_t)frame * n + t) * 96 + ch] = acc[r] + bv;
    }
  }
}

// ---------------------------------------------------------------------------
extern "C" void kernel_launch(void* const* d_in, const int* in_sizes, int n_in,
                              void* d_out, int out_size, void* d_ws, size_t ws_size,
                              hipStream_t stream) {
  (void)in_sizes; (void)n_in; (void)out_size; (void)ws_size;

  const float* x = (const float*)d_in[0];
  const float* kernel_r = (const float*)d_in[1];
  const float* kernel_i = (const float*)d_in[2];
  const float* bias_r = (const float*)d_in[3];
  float* out = (float*)d_out;
  float* ws = (float*)d_ws;

  const int NS[3] = {64, 128, 256};
  const int FE[3] = {33, 65, 129};
  const int FEP[3] = {48, 80, 144};
  const int MFR[3] = {1024, 512, 256};  // frames = B*T/n

  // workspace carve-up (floats)
  size_t off = 0;
  float *Wr[3], *Wi[3], *Cr[3], *Ci[3], *Kr[3], *Ki[3];
  for (int j = 0; j < 3; ++j) {
    size_t s = (size_t)FEP[j] * NS[j];
    Wr[j] = ws + off; off += s;
    Wi[j] = ws + off; off += s;
    Cr[j] = ws + off; off += s;
    Ci[j] = ws + off; off += s;
    Kr[j] = ws + off; off += (size_t)FEP[j] * 2048;
    Ki[j] = ws + off; off += (size_t)FEP[j] * 2048;
  }
  size_t omax = (size_t)1024 * 48 * 32;  // largest M*fep*32
  float* Ore = ws + off; off += omax;
  float* Oim = ws + off; off += omax;    // total ~17.8 MB of ws

  for (int j = 0; j < 3; ++j) {
    init_tables<<<dim3(256), dim3(256), 0, stream>>>(
        kernel_r, kernel_i, Wr[j], Wi[j], Cr[j], Ci[j], Kr[j], Ki[j],
        NS[j], FE[j], FEP[j]);
  }
  for (int j = 0; j < 3; ++j) {
    dim3 gab(FEP[j] / 16, MFR[j]);
    stage_ab<<<gab, dim3(128), 0, stream>>>(x, Wr[j], Wi[j], Kr[j], Ki[j],
                                            Ore, Oim, NS[j], FE[j], FEP[j]);
    stage_d<<<dim3(MFR[j]), dim3(256), 0, stream>>>(Cr[j], Ci[j], Ore, Oim,
                                                    bias_r, out, NS[j], FEP[j],
                                                    32 * j);
  }
}